// GATNetwork_257698038225
// MI455X (gfx1250) — compile-verified
//
#include <hip/hip_runtime.h>
#include <hip/hip_bf16.h>

typedef __attribute__((ext_vector_type(2))) float v2f;
typedef __attribute__((ext_vector_type(8))) float v8f;

#define NEG_SLOPE 0.01f
#define KSEG 25   // K-dimension segments for FC1 (occupancy for the 409.6MB stream)

__device__ __forceinline__ float leaky(float t) {
    return t >= 0.f ? t : NEG_SLOPE * t;
}

// ---------------- GCN part (rank-1 collapsed) ----------------

// deg=1 (self loop), s1=s2=0
__global__ void k_init(float* __restrict__ D, float* __restrict__ S1,
                       float* __restrict__ S2, int n) {
    int i = blockIdx.x * blockDim.x + threadIdx.x;
    if (i < n) { D[i] = 1.0f; S1[i] = 0.0f; S2[i] = 0.0f; }
}

// in-degree accumulation
__global__ void k_deg(float* __restrict__ D, const int* __restrict__ dst, int e) {
    int i = blockIdx.x * blockDim.x + threadIdx.x;
    if (i < e) atomicAdd(&D[dst[i]], 1.0f);
}

// D := rsqrt(deg);  A := x * dinv
__global__ void k_dinv(float* __restrict__ D, const float* __restrict__ x,
                       float* __restrict__ A, int n) {
    int i = blockIdx.x * blockDim.x + threadIdx.x;
    if (i < n) {
        float d = rsqrtf(D[i]);
        D[i] = d;
        A[i] = x[i] * d;
    }
}

// S[dst] += G[src]  (scalar message scatter; 1 atomic per edge)
__global__ void k_scatter(float* __restrict__ S, const int* __restrict__ src,
                          const int* __restrict__ dst, const float* __restrict__ G,
                          int e) {
    int i = blockIdx.x * blockDim.x + threadIdx.x;
    if (i < e) atomicAdd(&S[dst[i]], G[src[i]]);
}

// conv1 finalize + conv2 linear:  p = dinv*(s1 + a);
// g = sum_c leaky(p*W1c[c] + b1c[c]) * W2c[c];   S1 := g * dinv
__global__ void k_node1(const float* __restrict__ D, const float* __restrict__ A,
                        float* __restrict__ S1, const float* __restrict__ W1c,
                        const float* __restrict__ b1c, const float* __restrict__ W2c,
                        int conv, int n) {
    int i = blockIdx.x * blockDim.x + threadIdx.x;
    if (i >= n) return;
    float d = D[i];
    float p = d * (S1[i] + A[i]);
    float g = 0.f;
    for (int c = 0; c < conv; ++c) {
        float t = fmaf(p, W1c[c], b1c[c]);
        g = fmaf(leaky(t), W2c[c], g);
    }
    S1[i] = g * d;
}

// conv2 finalize: v = leaky(dinv*(s2 + ga) + b2c);  S2 := v
__global__ void k_node2(const float* __restrict__ D, const float* __restrict__ S1,
                        float* __restrict__ S2, const float* __restrict__ b2c, int n) {
    int i = blockIdx.x * blockDim.x + threadIdx.x;
    if (i >= n) return;
    float t = fmaf(D[i], S2[i] + S1[i], b2c[0]);
    S2[i] = leaky(t);
}

// ---------------- FC1: z = leaky(Wf1 @ v + bf1) via V_WMMA_F32_16X16X4_F32 ----
// Grid: (hid/16, KSEG). Block: 256 threads = 8 waves; each block owns 16 rows
// of Wf1 and one K-segment (~4000 of 100000). Wave w strides K with chunks of 4
// (k = k0 + w*4 + 32*t).
// A tile (16x4 f32): lane l -> row = l&15, K = (l>=16 ? 2 : 0) + {0,1}  (v2f)
// B tile (4x16 f32): column 0 carries v[k..k+3]; all lanes do the broadcast 8B
//   load, non-column-0 lanes zero it with a multiplicative mask (branchless).
// D column 0: lane 0 holds rows 0..7 in c[0..7], lane 16 holds rows 8..15.
// Per-(segment,row) partials go to Zpart; k_fc1_finish reduces + bias + leaky.
__global__ void __launch_bounds__(256)
k_fc1_wmma(const float* __restrict__ Wf1, const float* __restrict__ V,
           float* __restrict__ Zpart, int K, int hid) {
    __shared__ float red[8][16];
    const int lane = threadIdx.x & 31;
    const int wave = threadIdx.x >> 5;          // 0..7
    const int rowBase = blockIdx.x * 16;
    const int seg = blockIdx.y;

    const int segLen = (((K + KSEG - 1) / KSEG) + 15) & ~15;
    const int k0 = seg * segLen;
    const int k1 = min(K, k0 + segLen);
    const int span = (k1 > k0) ? (k1 - k0) : 0;
    const int Kmseg = k0 + (span & ~15);        // WMMA-covered K (tail scalar)

    const int row = rowBase + (lane & 15);
    const long long rowOff = (long long)row * (long long)K;
    const int hiK = (lane >> 4) ? 2 : 0;
    const float msk = ((lane & 15) == 0) ? 1.0f : 0.0f;  // keep only column 0 of B

    v8f c = {0.f, 0.f, 0.f, 0.f, 0.f, 0.f, 0.f, 0.f};

#pragma unroll 2
    for (int k = k0 + wave * 4; k < Kmseg; k += 32) {
        const int kl = k + hiK;
        v2f a = *(const v2f*)(Wf1 + rowOff + kl);
        __builtin_prefetch(Wf1 + rowOff + kl + 256, 0, 1);  // global_prefetch_b8
        v2f b = *(const v2f*)(V + kl);                      // broadcast load
        b.x *= msk;
        b.y *= msk;
        c = __builtin_amdgcn_wmma_f32_16x16x4_f32(
                /*neg_a=*/false, a, /*neg_b=*/false, b,
                /*c_mod=*/(short)0, c, /*reuse_a=*/false, /*reuse_b=*/false);
    }

    if ((lane & 15) == 0) {
        const int mBase = (lane >> 4) * 8;      // lane0 -> M 0..7, lane16 -> M 8..15
#pragma unroll
        for (int r = 0; r < 8; ++r) red[wave][mBase + r] = c[r];
    }
    __syncthreads();

    if (threadIdx.x < 16) {
        float acc = 0.f;
#pragma unroll
        for (int w = 0; w < 8; ++w) acc += red[w][threadIdx.x];
        const long long ro = (long long)(rowBase + threadIdx.x) * (long long)K;
        for (int k = Kmseg; k < k1; ++k)        // scalar tail (last segment only)
            acc = fmaf(Wf1[ro + k], V[k], acc);
        Zpart[(size_t)seg * hid + rowBase + threadIdx.x] = acc;
    }
}

// reduce KSEG partials, add bias, leaky
__global__ void k_fc1_finish(const float* __restrict__ Zpart,
                             const float* __restrict__ bf1,
                             float* __restrict__ Z, int hid) {
    int i = blockIdx.x * blockDim.x + threadIdx.x;
    if (i >= hid) return;
    float acc = bf1[i];
#pragma unroll 5
    for (int s = 0; s < KSEG; ++s) acc += Zpart[(size_t)s * hid + i];
    Z[i] = leaky(acc);
}

// ---------------- FC2: z2 = Wf2 @ z + bf2  (16 blocks, 16 rows each) ---------
// 256 threads = 16 output rows x 16 K-slices; LDS tree-reduce per row.
__global__ void __launch_bounds__(256)
k_fc2(const float* __restrict__ Wf2, const float* __restrict__ Z,
      const float* __restrict__ bf2, float* __restrict__ Z2, int hid) {
    __shared__ float red[256];
    const int t = threadIdx.x;
    const int o = blockIdx.x * 16 + (t >> 4);   // output row
    const int sub = t & 15;                     // K-slice id
    const int chunk = hid >> 4;                 // 64
    const float* wrow = Wf2 + (long long)o * hid + sub * chunk;
    const float* zp = Z + sub * chunk;

    float acc = 0.f;
    for (int k = 0; k < chunk; k += 4) {
        float4 w = *(const float4*)(wrow + k);
        float4 zv = *(const float4*)(zp + k);
        acc = fmaf(w.x, zv.x, acc);
        acc = fmaf(w.y, zv.y, acc);
        acc = fmaf(w.z, zv.z, acc);
        acc = fmaf(w.w, zv.w, acc);
    }
    red[t] = acc;
    __syncthreads();
    for (int s = 8; s > 0; s >>= 1) {
        if (sub < s) red[t] += red[t + s];
        __syncthreads();
    }
    if (sub == 0) Z2[o] = red[t] + bf2[o];
}

// ---------------- log_softmax over OUT logits (single tiny block) ------------
__global__ void __launch_bounds__(256)
k_logsoftmax(const float* __restrict__ Z2, float* __restrict__ out, int outn) {
    __shared__ float red[256];
    const int t = threadIdx.x;
    float v = Z2[t];

    red[t] = v;
    __syncthreads();
    for (int s = outn >> 1; s > 0; s >>= 1) {
        if (t < s) red[t] = fmaxf(red[t], red[t + s]);
        __syncthreads();
    }
    float mx = red[0];
    __syncthreads();

    red[t] = expf(v - mx);
    __syncthreads();
    for (int s = outn >> 1; s > 0; s >>= 1) {
        if (t < s) red[t] += red[t + s];
        __syncthreads();
    }
    out[t] = v - mx - logf(red[0]);
}

// ---------------- Launch ----------------
extern "C" void kernel_launch(void* const* d_in, const int* in_sizes, int n_in,
                              void* d_out, int out_size, void* d_ws, size_t ws_size,
                              hipStream_t stream) {
    const float* x   = (const float*)d_in[0];
    const int*   ei  = (const int*)  d_in[1];
    const float* W1c = (const float*)d_in[2];
    const float* b1c = (const float*)d_in[3];
    const float* W2c = (const float*)d_in[4];
    const float* b2c = (const float*)d_in[5];
    const float* Wf1 = (const float*)d_in[6];
    const float* bf1 = (const float*)d_in[7];
    const float* Wf2 = (const float*)d_in[8];
    const float* bf2 = (const float*)d_in[9];

    const int n    = in_sizes[0];       // N = 100000 (x is [N,1])
    const int e    = in_sizes[1] / 2;   // E = 3.2M
    const int conv = in_sizes[3];       // 16
    const int hid  = in_sizes[7];       // 1024
    const int outn = in_sizes[9];       // 256

    float* ws = (float*)d_ws;
    float* D     = ws;                    // deg -> dinv   [n]
    float* A     = ws + 1 * (size_t)n;    // x*dinv        [n]
    float* S1    = ws + 2 * (size_t)n;    // scatter1 -> ga[n]
    float* S2    = ws + 3 * (size_t)n;    // scatter2 -> v [n]
    float* Z     = ws + 4 * (size_t)n;    // hidden        [hid]
    float* Z2    = Z + hid;               // logits        [outn]
    float* Zpart = Z2 + outn;             // FC1 partials  [KSEG*hid]

    const int* src = ei;
    const int* dst = ei + e;

    const int nb = (n + 255) / 256;
    const int eb = (e + 255) / 256;

    k_init   <<<nb, 256, 0, stream>>>(D, S1, S2, n);
    k_deg    <<<eb, 256, 0, stream>>>(D, dst, e);
    k_dinv   <<<nb, 256, 0, stream>>>(D, x, A, n);
    k_scatter<<<eb, 256, 0, stream>>>(S1, src, dst, A, e);
    k_node1  <<<nb, 256, 0, stream>>>(D, A, S1, W1c, b1c, W2c, conv, n);
    k_scatter<<<eb, 256, 0, stream>>>(S2, src, dst, S1, e);
    k_node2  <<<nb, 256, 0, stream>>>(D, S1, S2, b2c, n);

    dim3 g1(hid / 16, KSEG);
    k_fc1_wmma  <<<g1, 256, 0, stream>>>(Wf1, S2, Zpart, n, hid);
    k_fc1_finish<<<(hid + 255) / 256, 256, 0, stream>>>(Zpart, bf1, Z, hid);
    k_fc2       <<<outn / 16, 256, 0, stream>>>(Wf2, Z, bf2, Z2, hid);
    k_logsoftmax<<<1, outn, 0, stream>>>(Z2, (float*)d_out, outn);
}